// DeformConv_51513837748708
// MI455X (gfx1250) — compile-verified
//
#include <hip/hip_runtime.h>
#include <hip/hip_bf16.h>

typedef __attribute__((ext_vector_type(2))) float v2f;
typedef __attribute__((ext_vector_type(8))) float v8f;

#define BATCH 8
#define CIN   128
#define COUT  128
#define HW    56
#define PIX   (HW*HW)          // 3136
#define NPIX  (BATCH*PIX)      // 25088
#define NTAP  9
#define HP    58               // padded extent
#define OFFC  (2*NTAP)         // 18
#define MOFF  32               // offset-conv M padded 18 -> 32
#define KC    64

// ---------------- Kernel 1a: reorder main conv weights -> Wr[tap][c][m] ----------------
__global__ __launch_bounds__(256)
void dcn_wreorder(const float* __restrict__ cw, float* __restrict__ wr) {
    int idx = blockIdx.x * blockDim.x + threadIdx.x;      // over NTAP*CIN*COUT
    int t = idx / (CIN * COUT);
    int c = (idx / COUT) % CIN;
    int m = idx % COUT;
    wr[idx] = cw[(m * CIN + c) * NTAP + t];
}

// ---------------- Kernel 1b: reorder offset conv weights -> Wo[tap][c][m<32] (zero-pad) ----
__global__ __launch_bounds__(256)
void dcn_woreorder(const float* __restrict__ ow, float* __restrict__ wo) {
    int idx = blockIdx.x * blockDim.x + threadIdx.x;      // over NTAP*CIN*MOFF
    int t = idx / (CIN * MOFF);
    int c = (idx / MOFF) % CIN;
    int m = idx % MOFF;
    wo[idx] = (m < OFFC) ? ow[(m * CIN + c) * NTAP + t] : 0.0f;
}

// ---------------- Kernel 2: offset conv as implicit WMMA GEMM ----------------
__global__ __launch_bounds__(256)
void dcn_offset_gemm(const float* __restrict__ x, const float* __restrict__ wo,
                     float* __restrict__ off) {
    __shared__ v2f As2[KC / 2][48];   // K-pair interleaved; cols 0..31 used; stride 48 (=16 mod 32)
    __shared__ v2f Bs2[KC / 2][80];   // cols 0..63 used; stride 80 (=16 mod 32)
    float* Asf = (float*)As2;
    float* Bsf = (float*)Bs2;

    const int tid  = threadIdx.x;
    const int wave = tid >> 5;
    const int lane = tid & 31;
    const int lm   = lane & 15;
    const int kh   = lane >> 4;
    const int mt   = wave & 1;        // M tile 0..1
    const int nt   = wave >> 1;       // N tile 0..3

    const int P0 = blockIdx.x * 64;
    const int bb = P0 / PIX;
    const int r0 = P0 % PIX;

    v8f acc = {};

    for (int tap = 0; tap < NTAP; ++tap) {
        const int dy = tap / 3 - 1, dx = tap % 3 - 1;
        for (int half = 0; half < 2; ++half) {
            const int c0 = half * KC;
#pragma unroll
            for (int i = 0; i < (KC * MOFF) / 256; ++i) {
                int e = tid + 256 * i;
                int kk = e >> 5, m = e & 31;
                Asf[(kk >> 1) * 96 + m * 2 + (kk & 1)] =
                    wo[(tap * CIN + c0 + kk) * MOFF + m];
            }
#pragma unroll
            for (int i = 0; i < (KC * 64) / 256; ++i) {
                int e = tid + 256 * i;
                int kk = e >> 6, j = e & 63;
                int r = r0 + j;
                int y = r / HW, xx = r % HW;
                int yy = y + dy, xq = xx + dx;
                float v = 0.0f;
                if (yy >= 0 && yy < HW && xq >= 0 && xq < HW)
                    v = x[(bb * CIN + c0 + kk) * PIX + yy * HW + xq];
                Bsf[(kk >> 1) * 160 + j * 2 + (kk & 1)] = v;
            }
            __syncthreads();
#pragma unroll
            for (int ks = 0; ks < KC / 4; ++ks) {
                v2f a = As2[2 * ks + kh][16 * mt + lm];
                v2f b = Bs2[2 * ks + kh][16 * nt + lm];
                acc = __builtin_amdgcn_wmma_f32_16x16x4_f32(
                    false, a, false, b, (short)0, acc, false, false);
            }
            __syncthreads();
        }
    }

    const int col = r0 + 16 * nt + lm;
#pragma unroll
    for (int j = 0; j < 8; ++j) {
        int m = 16 * mt + j + 8 * kh;
        if (m < OFFC)
            off[(bb * OFFC + m) * PIX + col] = acc[j];
    }
}

// ---------------- Kernel 3: bilinear sampling metadata per (pixel, tap) ----------------
__global__ __launch_bounds__(256)
void dcn_meta(const float* __restrict__ off, const float* __restrict__ ob,
              int4* __restrict__ midx, float4* __restrict__ mwt) {
    int idx = blockIdx.x * blockDim.x + threadIdx.x;      // over NPIX*NTAP
    int pix = idx / NTAP;
    int t   = idx % NTAP;
    int b   = pix / PIX;
    int r   = pix % PIX;
    int y   = r / HW, xx = r % HW;

    float offy = off[(b * OFFC + t)        * PIX + r] + ob[t];
    float offx = off[(b * OFFC + NTAP + t) * PIX + r] + ob[NTAP + t];
    float pny = (float)(t / 3) - 1.0f;
    float pnx = (float)(t % 3) - 1.0f;

    float py = (float)(y + 1)  + pny + offy;
    float px = (float)(xx + 1) + pnx + offx;

    float fy = floorf(py), fx = floorf(px);
    float lim = (float)(HP - 1);
    float qlt_y = fminf(fmaxf(fy, 0.0f), lim);
    float qlt_x = fminf(fmaxf(fx, 0.0f), lim);
    float qrb_y = fminf(fmaxf(fy + 1.0f, 0.0f), lim);
    float qrb_x = fminf(fmaxf(fx + 1.0f, 0.0f), lim);
    float pyc = fminf(fmaxf(py, 0.0f), lim);
    float pxc = fminf(fmaxf(px, 0.0f), lim);

    float g_lt = (1.0f - (pyc - qlt_y)) * truncf(1.0f - (pxc - qlt_x));
    float g_rb = (1.0f - (qrb_y - pyc)) * truncf(1.0f - (qrb_x - pxc));
    float g_lb = (1.0f - (pyc - qlt_y)) * (1.0f - (qrb_x - pxc));
    float g_rt = (1.0f - (qrb_y - pyc)) * (1.0f - (pxc - qlt_x));

    float qy[4] = { qlt_y, qrb_y, qlt_y, qrb_y };
    float qx[4] = { qlt_x, qrb_x, qrb_x, qlt_x };
    float gw[4] = { g_lt,  g_rb,  g_lb,  g_rt  };
    int   gi[4];
#pragma unroll
    for (int i = 0; i < 4; ++i) {
        int iy = (int)qy[i], ix = (int)qx[i];
        bool valid = (iy >= 1) && (iy <= HW) && (ix >= 1) && (ix <= HW);
        gi[i] = valid ? ((iy - 1) * HW + (ix - 1)) : 0;
        gw[i] = valid ? gw[i] : 0.0f;
    }
    midx[idx] = make_int4(gi[0], gi[1], gi[2], gi[3]);
    mwt[idx]  = make_float4(gw[0], gw[1], gw[2], gw[3]);
}

// ---------------- Kernel 4: main implicit GEMM with fused bilinear gather ----------------
__global__ __launch_bounds__(256)
void dcn_gemm(const float* __restrict__ x, const float* __restrict__ wr,
              const int4* __restrict__ midx, const float4* __restrict__ mwt,
              float* __restrict__ out) {
    __shared__ v2f As2[KC / 2][144];  // K-pair interleaved; cols 0..127 used (144 = 16 mod 32)
    __shared__ v2f Bs2[KC / 2][80];   // cols 0..63 used
    float* Asf = (float*)As2;
    float* Bsf = (float*)Bs2;

    const int tid  = threadIdx.x;
    const int wave = tid >> 5;
    const int lane = tid & 31;
    const int lm   = lane & 15;
    const int kh   = lane >> 4;

    const int P0 = blockIdx.x * 64;   // 3136 % 64 == 0: tile never crosses batch
    const int bb = P0 / PIX;
    const int r0 = P0 % PIX;

    v8f acc[4] = {};

    for (int tap = 0; tap < NTAP; ++tap) {
        for (int half = 0; half < 2; ++half) {
            const int c0 = half * KC;
#pragma unroll
            for (int i = 0; i < (KC * COUT) / 256; ++i) {
                int e = tid + 256 * i;
                int kk = e >> 7, m = e & 127;
                Asf[(kk >> 1) * 288 + m * 2 + (kk & 1)] =
                    wr[(tap * CIN + c0 + kk) * COUT + m];
            }
#pragma unroll
            for (int i = 0; i < (KC * 64) / 256; ++i) {
                int e = tid + 256 * i;
                int kk = e >> 6, j = e & 63;
                int pix = P0 + j;
                int4   mi = midx[pix * NTAP + tap];
                float4 mw = mwt[pix * NTAP + tap];
                const float* xb = x + ((bb * CIN + c0 + kk) * PIX);
                float v = mw.x * xb[mi.x];
                v = fmaf(mw.y, xb[mi.y], v);
                v = fmaf(mw.z, xb[mi.z], v);
                v = fmaf(mw.w, xb[mi.w], v);
                Bsf[(kk >> 1) * 160 + j * 2 + (kk & 1)] = v;
            }
            __syncthreads();

#pragma unroll
            for (int ks = 0; ks < KC / 4; ++ks) {
                v2f a = As2[2 * ks + kh][16 * wave + lm];
#pragma unroll
                for (int t = 0; t < 4; ++t) {
                    v2f b = Bs2[2 * ks + kh][16 * t + lm];
                    acc[t] = __builtin_amdgcn_wmma_f32_16x16x4_f32(
                        false, a, false, b, (short)0, acc[t], false, false);
                }
            }
            __syncthreads();
        }
    }

#pragma unroll
    for (int t = 0; t < 4; ++t) {
        int col = r0 + 16 * t + lm;
#pragma unroll
        for (int j = 0; j < 8; ++j) {
            int m = 16 * wave + j + 8 * kh;
            out[(bb * COUT + m) * PIX + col] = acc[t][j];
        }
    }
}

extern "C" void kernel_launch(void* const* d_in, const int* in_sizes, int n_in,
                              void* d_out, int out_size, void* d_ws, size_t ws_size,
                              hipStream_t stream) {
    const float* x  = (const float*)d_in[0];   // (8,128,56,56)
    const float* ow = (const float*)d_in[1];   // (18,128,3,3)
    const float* ob = (const float*)d_in[2];   // (18,)
    const float* cw = (const float*)d_in[3];   // (128,128,3,3)
    float* out = (float*)d_out;                // (8,128,56,56)

    char* ws = (char*)d_ws;
    const size_t OFF_BYTES  = (size_t)BATCH * OFFC * PIX * sizeof(float);     // 1.8 MB
    const size_t MIDX_BYTES = (size_t)NPIX * NTAP * sizeof(int4);             // 3.6 MB
    const size_t MWT_BYTES  = (size_t)NPIX * NTAP * sizeof(float4);           // 3.6 MB
    const size_t WR_BYTES   = (size_t)NTAP * CIN * COUT * sizeof(float);      // 0.59 MB
    float*  off  = (float*)ws;
    int4*   midx = (int4*)(ws + OFF_BYTES);
    float4* mwt  = (float4*)(ws + OFF_BYTES + MIDX_BYTES);
    float*  wrp  = (float*)(ws + OFF_BYTES + MIDX_BYTES + MWT_BYTES);
    float*  wop  = (float*)(ws + OFF_BYTES + MIDX_BYTES + MWT_BYTES + WR_BYTES);

    dcn_wreorder<<<(NTAP * CIN * COUT) / 256, 256, 0, stream>>>(cw, wrp);
    dcn_woreorder<<<(NTAP * CIN * MOFF) / 256, 256, 0, stream>>>(ow, wop);
    dcn_offset_gemm<<<NPIX / 64, 256, 0, stream>>>(x, wop, off);
    dcn_meta<<<(NPIX * NTAP) / 256, 256, 0, stream>>>(off, ob, midx, mwt);
    dcn_gemm<<<NPIX / 64, 256, 0, stream>>>(x, wrp, midx, mwt, out);
}